// FieldLayer_29411936043444
// MI455X (gfx1250) — compile-verified
//
#include <hip/hip_runtime.h>

// CDNA5 / gfx1250: wave32, V_WMMA_F32_16X16X4_F32.
typedef __attribute__((ext_vector_type(2))) float v2f;
typedef __attribute__((ext_vector_type(8))) float v8f;

// One wave per (b,v,t,n) site; there are b*v*t*n = 24576 sites.
//   x tile:  [h=16][k=8][f=8] f32 (4 KB; k=0,2 have zero weight and are never read)
//   angles:  dphi/dtheta [h=16][o=4]
//   output:  [o=4][f=8];  y[o,f] = (1/16) * sum_{h,k} enc[h,o,k] * x[h,k,f]
//   enc[h,o,:] = [0, 1, 0, 1, sin(.5*phi), cos(.5*phi), sin(.5*theta), cos(.5*theta)]
//
// WMMA f32 16x16x4 mapping (ISA 7.12.2): M=o (rows 0-3 live), N=f (cols 0-7 live),
// K marches over (h,k) in chunks of 4. A lane-half carries K={0,1} / K={2,3};
// B lane-half carries rows K={0,2} in vgpr0 halves and K={1,3} in vgpr1 halves.
__global__ __launch_bounds__(256) void field_layer_wmma_f32(
    const float* __restrict__ x,
    const float* __restrict__ dphi,
    const float* __restrict__ dtheta,
    float* __restrict__ out,
    int nsites)
{
    const int lane = threadIdx.x & 31;
    const int wave = threadIdx.x >> 5;
    const int site = blockIdx.x * 8 + wave;
    if (site >= nsites) return;            // wave-uniform: EXEC all-1s at every WMMA

    const float* __restrict__ xs = x      + (size_t)site * 1024; // [h][k][f]
    const float* __restrict__ ps = dphi   + (size_t)site * 64;   // [h][o]
    const float* __restrict__ ts = dtheta + (size_t)site * 64;   // [h][o]
    float*       __restrict__ os = out    + (size_t)site * 32;   // [o][f]

    const int   half = lane >> 4;          // 0: lanes 0-15, 1: lanes 16-31
    const int   l    = lane & 15;
    const int   fb   = l & 7;              // clamped f col (in-bounds for all lanes)
    const int   ob   = l & 3;              // clamped o row (in-bounds for all lanes)
    const float bm   = (l < 8) ? 1.0f : 0.0f;   // B column validity mask
    const float am   = (l < 4) ? 1.0f : 0.0f;   // A row validity mask

    // Per-lane base pointers: loop iterations differ only by an immediate offset,
    // so all loads are unconditional global_load_b32 with constant offsets.
    const float* __restrict__ bp1 = xs + (half ? 3 : 1) * 8 + fb;  // k=1 | k=3
    const float* __restrict__ bp4 = xs + (half ? 6 : 4) * 8 + fb;  // k=4 | k=6
    const float* __restrict__ bp5 = xs + (half ? 7 : 5) * 8 + fb;  // k=5 | k=7
    const float* __restrict__ ap  = (half ? ts : ps) + ob;         // phi | theta

    // Chunk-0 A matrix is constant: K slots enc k={0,2} -> 0, k={1,3} -> 1.
    v2f ac;
    ac.x = 0.0f;
    ac.y = am;

    v8f c = {};   // 16x16 f32 accumulator (rows 0-3 live)

#pragma unroll
    for (int h = 0; h < 16; ++h) {
        // ---- K-chunk 0: global k = 0..3 (k=0,2 weight zero -> x never loaded)
        v2f b0;
        b0.x = 0.0f;
        b0.y = bm * bp1[h * 64];
        c = __builtin_amdgcn_wmma_f32_16x16x4_f32(false, ac, false, b0,
                                                  (short)0, c, false, false);

        // ---- K-chunk 1: global k = 4..7 -> sin/cos(.5*phi) | sin/cos(.5*theta)
        const float ang = 0.5f * ap[h * 4];
        v2f a1;
        a1.x = am * __sinf(ang);           // k=4 | k=6
        a1.y = am * __cosf(ang);           // k=5 | k=7
        v2f b1;
        b1.x = bm * bp4[h * 64];
        b1.y = bm * bp5[h * 64];
        c = __builtin_amdgcn_wmma_f32_16x16x4_f32(false, a1, false, b1,
                                                  (short)0, c, false, false);
    }

    // D layout: VGPR r = row M=r for lanes 0-15, N = lane. y[o][f] = c[o], lanes 0-7.
    if (lane < 8) {
        os[0 * 8 + lane] = c[0] * 0.0625f;
        os[1 * 8 + lane] = c[1] * 0.0625f;
        os[2 * 8 + lane] = c[2] * 0.0625f;
        os[3 * 8 + lane] = c[3] * 0.0625f;
    }
}

extern "C" void kernel_launch(void* const* d_in, const int* in_sizes, int n_in,
                              void* d_out, int out_size, void* d_ws, size_t ws_size,
                              hipStream_t stream) {
    const float* x      = (const float*)d_in[0];
    const float* dphi   = (const float*)d_in[1];
    const float* dtheta = (const float*)d_in[2];
    float* out = (float*)d_out;

    const int nsites = in_sizes[0] / 1024;      // x: nh*c = 16*64 floats per site
    const int blocks = (nsites + 7) / 8;        // 8 waves (sites) per 256-thread block
    field_layer_wmma_f32<<<blocks, 256, 0, stream>>>(x, dphi, dtheta, out, nsites);
}